// Gaussians_46600395161904
// MI455X (gfx1250) — compile-verified
//
#include <hip/hip_runtime.h>
#include <hip/hip_bf16.h>
#include <math.h>

typedef __attribute__((ext_vector_type(2))) float v2f;
typedef __attribute__((ext_vector_type(8))) float v8f;

namespace {
constexpr int   kN             = 12288;
constexpr int   kTile          = 16;
constexpr int   kTiles         = kN / kTile;              // 768
constexpr int   kWavesPerBlock = 4;
constexpr int   kBlockThreads  = kWavesPerBlock * 32;     // 128
constexpr int   kBlocks        = kTiles / kWavesPerBlock; // 192
constexpr float kBaseScale     = 0.001f;
constexpr float kMinScale      = 1e-5f;
constexpr int   kInfBits       = 0x7f800000;              // +INF bit pattern
}

// Pack points as (x, y, z, |p|^2): one 16B record feeds A, B and C-init.
__global__ void pack_points_kernel(const float* __restrict__ pts,
                                   float4* __restrict__ p4, int n) {
  int i = blockIdx.x * blockDim.x + threadIdx.x;
  if (i < n) {
    float x = pts[3 * i + 0];
    float y = pts[3 * i + 1];
    float z = pts[3 * i + 2];
    p4[i] = make_float4(x, y, z, x * x + y * y + z * z);
  }
}

__device__ __forceinline__ int imin(int a, int b) { return a < b ? a : b; }
__device__ __forceinline__ int imax(int a, int b) { return a > b ? a : b; }

// Branchless ascending top-3 insert on f32 bit patterns (v_min_i32/v_max_i32,
// no NaN canonicalization). Valid because every tracked value is either +INF
// or a d^2 whose sign-bit cases all clamp to zero distance at the end.
__device__ __forceinline__ void top3_insert_i(int v, int& t0, int& t1,
                                              int& t2) {
  int m;
  m = imin(t0, v); v = imax(t0, v); t0 = m;
  m = imin(t1, v); v = imax(t1, v); t1 = m;
  t2 = imin(t2, v);
}

// One 16x16 d^2 tile from an already-loaded A fragment: one
// v_wmma_f32_16x16x4_f32, v_min3_i32 tree, wave-uniform early-out insert.
__device__ __forceinline__ void tile_compute(v2f a, v2f b, const v8f& cinit,
                                             int& t0, int& t1, int& t2) {
  v8f d = __builtin_amdgcn_wmma_f32_16x16x4_f32(
      /*neg_a=*/false, a, /*neg_b=*/false, b,
      /*c_mod=*/(short)0, cinit, /*reuse_a=*/false, /*reuse_b=*/false);

  int v0 = __float_as_int(d[0]);
  int v1 = __float_as_int(d[1]);
  int v2 = __float_as_int(d[2]);
  int v3 = __float_as_int(d[3]);
  int v4 = __float_as_int(d[4]);
  int v5 = __float_as_int(d[5]);
  int v6 = __float_as_int(d[6]);
  int v7 = __float_as_int(d[7]);

  int m8 = imin(imin(imin(v0, v1), imin(v2, v3)),
                imin(imin(v4, v5), imin(v6, v7)));
  if (__any(m8 < t2)) {  // wave-uniform skip of the expensive insert
    top3_insert_i(v0, t0, t1, t2);
    top3_insert_i(v1, t0, t1, t2);
    top3_insert_i(v2, t0, t1, t2);
    top3_insert_i(v3, t0, t1, t2);
    top3_insert_i(v4, t0, t1, t2);
    top3_insert_i(v5, t0, t1, t2);
    top3_insert_i(v6, t0, t1, t2);
    top3_insert_i(v7, t0, t1, t2);
  }
}

// Sweep `cnt` consecutive tiles starting at `base` (this lane's component of
// the first tile; tiles stride 64 floats).  Software pipeline with 4 tiles in
// flight: the 4 loads for the next group are issued before the 4 wmma+select
// bodies of the current group, so global_load latency is covered by ~50
// instructions instead of being exposed behind an s_wait_loadcnt 0.
__device__ __forceinline__ void sweep_range(const float* __restrict__ base,
                                            int cnt, v2f b, const v8f& cinit,
                                            int& t0, int& t1, int& t2) {
  auto ld = [&](int i) {
    return *reinterpret_cast<const v2f*>(base + 64 * i);
  };
  int i = 0;
  if (cnt >= 4) {
    v2f a0 = ld(0), a1 = ld(1), a2 = ld(2), a3 = ld(3);
    for (; i + 8 <= cnt; i += 4) {
      v2f n0 = ld(i + 4), n1 = ld(i + 5), n2 = ld(i + 6), n3 = ld(i + 7);
      tile_compute(a0, b, cinit, t0, t1, t2);
      tile_compute(a1, b, cinit, t0, t1, t2);
      tile_compute(a2, b, cinit, t0, t1, t2);
      tile_compute(a3, b, cinit, t0, t1, t2);
      a0 = n0; a1 = n1; a2 = n2; a3 = n3;
    }
    tile_compute(a0, b, cinit, t0, t1, t2);
    tile_compute(a1, b, cinit, t0, t1, t2);
    tile_compute(a2, b, cinit, t0, t1, t2);
    tile_compute(a3, b, cinit, t0, t1, t2);
    i += 4;
  }
  for (; i < cnt; ++i) tile_compute(ld(i), b, cinit, t0, t1, t2);
}

// One wave = 16 query points.  d^2 tile via V_WMMA_F32_16X16X4_F32:
//   A[m] = ( x_m,  y_m,  z_m, |p_m|^2 )   (neighbors, SRC0)
//   B[n] = (-2x_n, -2y_n, -2z_n, 1 )      (queries,   SRC1)
//   C[m][n] = |p_n|^2  (+INF at the self-distance element of tile qb)
// => D[m][n] = |p_m|^2 + |p_n|^2 - 2 p_m.p_n
__global__ __launch_bounds__(kBlockThreads) void knn_cov_kernel(
    const float4* __restrict__ p4, const float4* __restrict__ quat,
    float* __restrict__ out) {
  const int  lane = threadIdx.x & 31;
  const int  wave = threadIdx.x >> 5;
  const int  qb   = blockIdx.x * kWavesPerBlock + wave;
  const int  lo   = lane & 15;
  const bool hi   = lane >= 16;
  const int  n    = qb * kTile + lo;

  // B fragment (4x16 f32): lanes 0-15 hold K=0,1; lanes 16-31 hold K=2,3.
  const float4 qp = p4[n];
  v2f b;
  if (!hi) { b[0] = -2.0f * qp.x; b[1] = -2.0f * qp.y; }
  else     { b[0] = -2.0f * qp.z; b[1] = 1.0f; }
  const float sqn = qp.w;

  // C-init fragments: plain (|p_n|^2 broadcast) and diagonal-poisoned.
  v8f c_plain;
#pragma unroll
  for (int k = 0; k < 8; ++k) c_plain[k] = sqn;
  const int kd = lo - (hi ? 8 : 0);  // diagonal k for this lane (may be OOR)
  v8f       c_diag;
#pragma unroll
  for (int k = 0; k < 8; ++k)
    c_diag[k] = (k == kd) ? __builtin_inff() : c_plain[k];

  const float* pf    = reinterpret_cast<const float*>(p4);
  const int    comp  = hi ? 2 : 0;  // lane<16 reads (x,y); lane>=16 (z,|p|^2)
  const float* base0 = pf + 4 * lo + comp;  // this lane's slice of tile 0

  int t0 = kInfBits, t1 = kInfBits, t2 = kInfBits;

  // Three-phase sweep: diagonal handling costs zero in the hot loops.
  sweep_range(base0, qb, b, c_plain, t0, t1, t2);  // tiles [0, qb)
  tile_compute(*reinterpret_cast<const v2f*>(base0 + 64 * qb), b, c_diag, t0,
               t1, t2);                            // tile qb (diagonal)
  sweep_range(base0 + 64 * (qb + 1), kTiles - qb - 1, b, c_plain, t0, t1,
              t2);                                 // tiles (qb, kTiles)

  // Merge the two half-wave row groups (lane L and L+16 share a query).
  int u0 = __shfl_xor(t0, 16, 32);
  int u1 = __shfl_xor(t1, 16, 32);
  int u2 = __shfl_xor(t2, 16, 32);
  top3_insert_i(u0, t0, t1, t2);
  top3_insert_i(u1, t0, t1, t2);
  top3_insert_i(u2, t0, t1, t2);

  if (!hi) {
    float d0 = sqrtf(fmaxf(__int_as_float(t0), 0.0f));
    float d1 = sqrtf(fmaxf(__int_as_float(t1), 0.0f));
    float d2 = sqrtf(fmaxf(__int_as_float(t2), 0.0f));
    float s  = kBaseScale * fmaxf((d0 + d1 + d2) * (1.0f / 3.0f), kMinScale);

    // quaternion (w,x,y,z) -> M = s*R -> cov = M M^T (symmetric)
    const float4 q   = quat[n];
    float        r   = q.x, x = q.y, y = q.z, z = q.w;
    float        inv = 1.0f / sqrtf(r * r + x * x + y * y + z * z);
    r *= inv; x *= inv; y *= inv; z *= inv;

    float M00 = s * (1.0f - 2.0f * (y * y + z * z));
    float M01 = s * (2.0f * (x * y - r * z));
    float M02 = s * (2.0f * (x * z + r * y));
    float M10 = s * (2.0f * (x * y + r * z));
    float M11 = s * (1.0f - 2.0f * (x * x + z * z));
    float M12 = s * (2.0f * (y * z - r * x));
    float M20 = s * (2.0f * (x * z - r * y));
    float M21 = s * (2.0f * (y * z + r * x));
    float M22 = s * (1.0f - 2.0f * (x * x + y * y));

    float c00 = M00 * M00 + M01 * M01 + M02 * M02;
    float c01 = M00 * M10 + M01 * M11 + M02 * M12;
    float c02 = M00 * M20 + M01 * M21 + M02 * M22;
    float c11 = M10 * M10 + M11 * M11 + M12 * M12;
    float c12 = M10 * M20 + M11 * M21 + M12 * M22;
    float c22 = M20 * M20 + M21 * M21 + M22 * M22;

    float* o = out + n * 9;
    o[0] = c00; o[1] = c01; o[2] = c02;
    o[3] = c01; o[4] = c11; o[5] = c12;
    o[6] = c02; o[7] = c12; o[8] = c22;
  }
}

extern "C" void kernel_launch(void* const* d_in, const int* in_sizes, int n_in,
                              void* d_out, int out_size, void* d_ws,
                              size_t ws_size, hipStream_t stream) {
  const float*  pts  = (const float*)d_in[0];   // [N,3] f32
  const float4* quat = (const float4*)d_in[1];  // [N,4] f32
  float*        out  = (float*)d_out;           // [N,3,3] f32
  float4*       p4   = (float4*)d_ws;           // N * 16 B = 192 KB scratch

  pack_points_kernel<<<(kN + 255) / 256, 256, 0, stream>>>(pts, p4, kN);
  knn_cov_kernel<<<kBlocks, kBlockThreads, 0, stream>>>(p4, quat, out);
}